// ImprovedGraphSAGE_58042188038363
// MI455X (gfx1250) — compile-verified
//
#include <hip/hip_runtime.h>
#include <hip/hip_bf16.h>

#define N_NODES 100000
#define N_EDGES 1600000
#define IN_CH   165
#define IN_PAD  168      // IN_CH padded to a multiple of 4 (K-dim of f32 WMMA)
#define HID     128

typedef __attribute__((ext_vector_type(2))) float v2f;
typedef __attribute__((ext_vector_type(8))) float v8f;

// ---------------------------------------------------------------- utilities
__global__ void zero_f32_kernel(float* __restrict__ p, size_t n) {
    size_t i = (size_t)blockIdx.x * blockDim.x + threadIdx.x;
    if (i < n) p[i] = 0.0f;
}

// dst[rows x dstD] = src[rows x srcD] zero-padded on the right
__global__ void pad_rows_kernel(const float* __restrict__ src,
                                float* __restrict__ dst,
                                int srcD, int dstD, long rows) {
    long i = (long)blockIdx.x * blockDim.x + threadIdx.x;
    if (i >= rows * dstD) return;
    int col  = (int)(i % dstD);
    long row = i / dstD;
    dst[i] = (col < srcD) ? src[row * srcD + col] : 0.0f;
}

__global__ void count_kernel(const int* __restrict__ dst,
                             float* __restrict__ cnt, int nEdges) {
    int e = blockIdx.x * blockDim.x + threadIdx.x;
    if (e < nEdges) atomicAdd(&cnt[dst[e]], 1.0f);
}

__global__ void rdeg_kernel(const float* __restrict__ cnt,
                            float* __restrict__ rdeg, int n) {
    int i = blockIdx.x * blockDim.x + threadIdx.x;
    if (i < n) rdeg[i] = 1.0f / fmaxf(cnt[i], 1.0f);
}

// ------------------------------------------------- edge scatter (mean-agg numerator)
// one wave per edge; lanes stride the feature dimension
__global__ __launch_bounds__(256)
void scatter_kernel(const float* __restrict__ h,
                    const int* __restrict__ src,
                    const int* __restrict__ dst,
                    float* __restrict__ agg,
                    int d, int nEdges) {
    int e = blockIdx.x * (blockDim.x >> 5) + (threadIdx.x >> 5);
    if (e >= nEdges) return;
    int lane = threadIdx.x & 31;
    int s = src[e];
    int t = dst[e];
    const float* __restrict__ hs = h + (size_t)s * d;
    float* __restrict__ at = agg + (size_t)t * d;
    for (int f = lane; f < d; f += 32)
        atomicAdd(&at[f], hs[f]);
}

// ------------------------------------------------- fused SAGE GEMM via f32 WMMA
// out[N x 128] = act( (agg .* rdeg) @ Wl^T + h @ Wr^T + b )
// one wave per 16x16 output tile; 8 waves/block cover all 128 output columns
// for one 16-node row block. D is compile-time (multiple of 4) -> branchless
// float2 (b64) loads, no EXEC divergence around the WMMAs.
template <int D, bool RELU>
__global__ __launch_bounds__(256)
void sage_gemm_wmma_kernel(const float* __restrict__ agg,
                           const float* __restrict__ rdeg,
                           const float* __restrict__ h,
                           const float* __restrict__ Wl,   // [128 x D] row-major
                           const float* __restrict__ Wr,   // [128 x D] row-major
                           const float* __restrict__ bias, // [128]
                           float* __restrict__ out) {      // [N x 128]
    static_assert(D % 4 == 0, "K must be a multiple of 4");
    const int lane = threadIdx.x & 31;
    const int wave = threadIdx.x >> 5;          // 0..7 -> column tile
    const int m0   = blockIdx.x * 16;           // node row block
    const int n0   = wave * 16;

    const int mrow = m0 + (lane & 15);          // A row for this lane
    const int nrow = n0 + (lane & 15);          // B column (== weight row)
    const int koff = (lane >> 4) << 1;          // 0 (lanes 0-15) or 2 (lanes 16-31)

    const float rd = rdeg[mrow];

    // float2 views at this lane's K phase; rows are (D*4)B with D%4==0 and
    // koff even, so every access is 8-byte aligned -> global_load_b64.
    const float2* __restrict__ arow  = (const float2*)(agg + (size_t)mrow * D + koff);
    const float2* __restrict__ hrow  = (const float2*)(h   + (size_t)mrow * D + koff);
    const float2* __restrict__ wlrow = (const float2*)(Wl  + (size_t)nrow * D + koff);
    const float2* __restrict__ wrrow = (const float2*)(Wr  + (size_t)nrow * D + koff);

    v8f c = {0.f, 0.f, 0.f, 0.f, 0.f, 0.f, 0.f, 0.f};

#pragma unroll 4
    for (int k = 0; k < D / 4; ++k) {
        float2 am = arow[2 * k];    // floats (4k+koff, 4k+koff+1) of agg row
        float2 bl = wlrow[2 * k];
        float2 ah = hrow[2 * k];
        float2 br = wrrow[2 * k];
        v2f a1; a1.x = am.x * rd; a1.y = am.y * rd;
        v2f b1; b1.x = bl.x;      b1.y = bl.y;
        v2f a2; a2.x = ah.x;      a2.y = ah.y;
        v2f b2; b2.x = br.x;      b2.y = br.y;
        c = __builtin_amdgcn_wmma_f32_16x16x4_f32(
                false, a1, false, b1, (short)0, c, false, false);
        c = __builtin_amdgcn_wmma_f32_16x16x4_f32(
                false, a2, false, b2, (short)0, c, false, false);
    }

    // C/D layout: VGPR r -> M = m0 + r + (lane>=16 ? 8 : 0), N = n0 + (lane&15)
    const float bv = bias[nrow];
    const int mbase = m0 + ((lane < 16) ? 0 : 8);
#pragma unroll
    for (int r = 0; r < 8; ++r) {
        float v = c[r] + bv;
        if (RELU) v = fmaxf(v, 0.f);
        out[(size_t)(mbase + r) * HID + nrow] = v;
    }
}

// ------------------------------------------------- tiny output head [N,2]
__global__ __launch_bounds__(256)
void out_head_kernel(const float* __restrict__ h,
                     const float* __restrict__ Wout,  // [2 x 128]
                     const float* __restrict__ bout,  // [2]
                     float* __restrict__ out, int n) {
    int i = blockIdx.x * blockDim.x + threadIdx.x;
    if (i >= n) return;
    const float4* hr = (const float4*)(h + (size_t)i * HID);
    const float4* w0 = (const float4*)(Wout);
    const float4* w1 = (const float4*)(Wout + HID);
    float a0 = bout[0], a1 = bout[1];
#pragma unroll 8
    for (int k = 0; k < HID / 4; ++k) {
        float4 hv = hr[k];
        float4 u = w0[k];
        float4 v = w1[k];
        a0 += hv.x * u.x + hv.y * u.y + hv.z * u.z + hv.w * u.w;
        a1 += hv.x * v.x + hv.y * v.y + hv.z * v.z + hv.w * v.w;
    }
    out[(size_t)i * 2 + 0] = a0;
    out[(size_t)i * 2 + 1] = a1;
}

// ---------------------------------------------------------------- launcher
extern "C" void kernel_launch(void* const* d_in, const int* in_sizes, int n_in,
                              void* d_out, int out_size, void* d_ws, size_t ws_size,
                              hipStream_t stream) {
    const float* x    = (const float*)d_in[0];
    const int*   ei   = (const int*)d_in[1];
    const float* Wl0  = (const float*)d_in[2];
    const float* bl0  = (const float*)d_in[3];
    const float* Wr0  = (const float*)d_in[4];
    const float* Wl1  = (const float*)d_in[5];
    const float* bl1  = (const float*)d_in[6];
    const float* Wr1  = (const float*)d_in[7];
    const float* Wl2  = (const float*)d_in[8];
    const float* bl2  = (const float*)d_in[9];
    const float* Wr2  = (const float*)d_in[10];
    const float* Wout = (const float*)d_in[11];
    const float* bout = (const float*)d_in[12];
    float* out = (float*)d_out;

    const int* src = ei;
    const int* dst = ei + N_EDGES;

    // workspace layout (floats); hB aliases xp (xp dead after layer-0 GEMM)
    float* ws   = (float*)d_ws;
    float* cnt  = ws;                                   // N
    float* rdeg = cnt + N_NODES;                        // N
    float* agg  = rdeg + N_NODES;                       // N * IN_PAD
    float* hA   = agg + (size_t)N_NODES * IN_PAD;       // N * HID
    float* xp   = hA + (size_t)N_NODES * HID;           // N * IN_PAD
    float* hB   = xp;                                   // N * HID (alias)
    float* Wl0p = xp + (size_t)N_NODES * IN_PAD;        // 128 * IN_PAD
    float* Wr0p = Wl0p + (size_t)HID * IN_PAD;          // 128 * IN_PAD

    const int ZT = 256;
    auto zero = [&](float* p, size_t n) {
        zero_f32_kernel<<<(unsigned)((n + ZT - 1) / ZT), ZT, 0, stream>>>(p, n);
    };
    auto pad = [&](const float* s, float* dptr, int sd, int dd, long rows) {
        long n = rows * dd;
        pad_rows_kernel<<<(unsigned)((n + ZT - 1) / ZT), ZT, 0, stream>>>(s, dptr, sd, dd, rows);
    };

    const int edgeBlocks256 = (N_EDGES + 255) / 256;
    const int scatBlocks    = (N_EDGES + 7) / 8;     // 8 waves per 256-thread block
    const int gemmBlocks    = N_NODES / 16;          // 6250
    const int nodeBlocks256 = (N_NODES + 255) / 256;

    // degrees (shared by all layers)
    zero(cnt, (size_t)N_NODES);
    count_kernel<<<edgeBlocks256, 256, 0, stream>>>(dst, cnt, N_EDGES);
    rdeg_kernel<<<nodeBlocks256, 256, 0, stream>>>(cnt, rdeg, N_NODES);

    // pad layer-0 operands to K=168
    pad(x,   xp,   IN_CH, IN_PAD, N_NODES);
    pad(Wl0, Wl0p, IN_CH, IN_PAD, HID);
    pad(Wr0, Wr0p, IN_CH, IN_PAD, HID);

    // ---- layer 0: xp (K=168) -> hA, ReLU
    zero(agg, (size_t)N_NODES * IN_PAD);
    scatter_kernel<<<scatBlocks, 256, 0, stream>>>(xp, src, dst, agg, IN_PAD, N_EDGES);
    sage_gemm_wmma_kernel<IN_PAD, true><<<gemmBlocks, 256, 0, stream>>>(
        agg, rdeg, xp, Wl0p, Wr0p, bl0, hA);

    // ---- layer 1: hA (K=128) -> hB, ReLU
    zero(agg, (size_t)N_NODES * HID);
    scatter_kernel<<<scatBlocks, 256, 0, stream>>>(hA, src, dst, agg, HID, N_EDGES);
    sage_gemm_wmma_kernel<HID, true><<<gemmBlocks, 256, 0, stream>>>(
        agg, rdeg, hA, Wl1, Wr1, bl1, hB);

    // ---- layer 2: hB (K=128) -> hA, no ReLU
    zero(agg, (size_t)N_NODES * HID);
    scatter_kernel<<<scatBlocks, 256, 0, stream>>>(hB, src, dst, agg, HID, N_EDGES);
    sage_gemm_wmma_kernel<HID, false><<<gemmBlocks, 256, 0, stream>>>(
        agg, rdeg, hB, Wl2, Wr2, bl2, hA);

    // ---- output head
    out_head_kernel<<<nodeBlocks256, 256, 0, stream>>>(hA, Wout, bout, out, N_NODES);
}